// GCN_50405736186128
// MI455X (gfx1250) — compile-verified
//
#include <hip/hip_runtime.h>

// GCN, 3 layers: N=50000 nodes, E=800000 edges, dims 512 -> 128 -> 128 -> 16.
// fp32 end-to-end using V_WMMA_F32_16X16X4_F32 for the dense GEMMs.

#define NN 50000
#define NE 800000

typedef __attribute__((ext_vector_type(2))) float v2f;
typedef __attribute__((ext_vector_type(4))) float v4f;
typedef __attribute__((ext_vector_type(8))) float v8f;

// ---------------- degree / normalization ----------------

__global__ void k_fill_ones(float* __restrict__ p, int n) {
    int i = blockIdx.x * blockDim.x + threadIdx.x;
    if (i < n) p[i] = 1.0f;   // self-loop contributes 1 to every node
}

__global__ void k_deg_accum(const long long* __restrict__ dst,
                            float* __restrict__ deg, int e) {
    int i = blockIdx.x * blockDim.x + threadIdx.x;
    if (i < e) atomicAdd(&deg[(int)dst[i]], 1.0f);
}

__global__ void k_deg_rsqrt(float* __restrict__ p, int n) {
    int i = blockIdx.x * blockDim.x + threadIdx.x;
    if (i < n) p[i] = rsqrtf(p[i]);   // deg >= 1 always (self loops)
}

// ---------------- WMMA fp32 GEMM: H = (relu?)X[nrows x K_IN] * W[K_IN x N_OUT] ----------------
// Block: 256 threads = 8 waves; block tile = 128 rows x N_OUT cols.
// Wave w owns rows [blockRow + 16w, +16) and all N_OUT columns (N_OUT/16 accum tiles).
// K chunked by 32 through LDS.
//   Xs: row stride 36 dwords -> A-frag b64 loads hit disjoint banks per half-wave.
//   Ws: K-pair interleaved [kpair][col*2 + (k&1)] so a B-frag is ONE aligned ds_load_b64
//       (v2f straight into an even VGPR pair -> no v_mov shuffles before v_wmma).
//       Pair-row stride ≡ 32 (mod 64) dwords -> half-waves on disjoint bank halves.

template <int K_IN, int N_OUT, bool RELU>
__global__ __launch_bounds__(256)
void k_gcn_gemm(const float* __restrict__ X, const float* __restrict__ W,
                float* __restrict__ H, int nrows) {
    constexpr int KC = 32;
    constexpr int XS = KC + 4;                        // 36
    constexpr int WPAIR = KC / 2;                     // 16 pair-rows
    constexpr int WS2 = (N_OUT == 16) ? 96 : (2 * N_OUT + 32); // 96 / 288 : ≡32 mod 64
    constexpr int NT = N_OUT / 16;

    __shared__ float Xs[128][XS];
    __shared__ float Ws[WPAIR][WS2];

    const int tid  = threadIdx.x;
    const int lane = tid & 31;
    const int wave = tid >> 5;
    const int m    = lane & 15;          // row within A tile / col within B,C tile
    const int kh   = (lane >> 4) << 1;   // 0 or 2: K phase of this half-wave
    const int rowBase = blockIdx.x * 128;
    const int wrow    = rowBase + wave * 16;

    v8f acc[NT] = {};

    for (int k0 = 0; k0 < K_IN; k0 += KC) {
        // cooperative load of X tile (128 x 32), relu folded in, OOB rows -> 0
        for (int i = tid; i < 128 * KC; i += 256) {
            int r = i / KC, c = i % KC;
            int gr = rowBase + r;
            float v = 0.0f;
            if (gr < nrows) {
                v = X[gr * K_IN + k0 + c];
                if (RELU) v = fmaxf(v, 0.0f);
            }
            Xs[r][c] = v;
        }
        // cooperative load of W tile (32 x N_OUT), K-pair interleaved
        for (int i = tid; i < KC * N_OUT; i += 256) {
            int r = i / N_OUT, c = i % N_OUT;
            Ws[r >> 1][c * 2 + (r & 1)] = W[(k0 + r) * N_OUT + c];
        }
        // prefetch next X chunk into the cache hierarchy (global_prefetch_b8)
        if (k0 + KC < K_IN) {
            int pr = rowBase + (tid >> 1);
            if (pr < nrows)
                __builtin_prefetch(&X[pr * K_IN + k0 + KC + (tid & 1) * 16], 0, 1);
        }
        __syncthreads();

#pragma unroll
        for (int kk = 0; kk < KC; kk += 4) {
            // A frag: lane holds A[m][kk+kh .. kk+kh+1]  (one b64 LDS load)
            const v2f a = *(const v2f*)&Xs[wave * 16 + m][kk + kh];
            const int prow = (kk + kh) >> 1;
#pragma unroll
            for (int t = 0; t < NT; ++t) {
                // B frag: lane holds B[kk+kh][col], B[kk+kh+1][col] (one b64 LDS load)
                const v2f b = *(const v2f*)&Ws[prow][(t * 16 + m) * 2];
                acc[t] = __builtin_amdgcn_wmma_f32_16x16x4_f32(
                    false, a, false, b, (short)0, acc[t], false, false);
            }
        }
        __syncthreads();
    }

    // C/D layout: vgpr i, lanes 0-15 -> M=i ; lanes 16-31 -> M=8+i ; N=lane%16
    if (wrow < nrows) {   // nrows % 16 == 0 -> whole wave tile in/out
        const int roff = (lane >> 4) * 8;
#pragma unroll
        for (int t = 0; t < NT; ++t) {
#pragma unroll
            for (int i = 0; i < 8; ++i) {
                H[(wrow + roff + i) * N_OUT + t * 16 + m] = acc[t][i];
            }
        }
    }
}

// ---------------- aggregation ----------------

// out[v,f] = dinv[v]^2 * h[v,f] + bias[f]   (self-loop term + bias), float4-vectorized
template <int F>
__global__ void k_agg_init(const float* __restrict__ h, const float* __restrict__ dinv,
                           const float* __restrict__ bias, float* __restrict__ out, int n) {
    int i = blockIdx.x * blockDim.x + threadIdx.x;      // indexes float4s
    if (i < n * (F / 4)) {
        int v  = i / (F / 4);
        int f4 = i % (F / 4);
        float d  = dinv[v];
        float d2 = d * d;
        const v4f hv = *(const v4f*)&h[v * F + f4 * 4];
        const v4f bv = *(const v4f*)&bias[f4 * 4];
        v4f o = d2 * hv + bv;
        *(v4f*)&out[v * F + f4 * 4] = o;
    }
}

// out[dst] += dinv[src]*dinv[dst] * h[src]  — LPE lanes per edge, 4 floats per lane
template <int F, int LPE>
__global__ void k_agg_edges(const long long* __restrict__ src,
                            const long long* __restrict__ dst,
                            const float* __restrict__ dinv,
                            const float* __restrict__ h,
                            float* __restrict__ out, int e) {
    int gid = blockIdx.x * blockDim.x + threadIdx.x;
    int ei = gid / LPE;
    if (ei >= e) return;
    int sub = gid % LPE;                 // F == LPE*4
    int s = (int)src[ei];
    int d = (int)dst[ei];
    float w = dinv[s] * dinv[d];
    const float4 hv = *(const float4*)&h[s * F + sub * 4];
    float* op = &out[d * F + sub * 4];
    atomicAdd(op + 0, w * hv.x);
    atomicAdd(op + 1, w * hv.y);
    atomicAdd(op + 2, w * hv.z);
    atomicAdd(op + 3, w * hv.w);
}

// ---------------- launcher ----------------

extern "C" void kernel_launch(void* const* d_in, const int* in_sizes, int n_in,
                              void* d_out, int out_size, void* d_ws, size_t ws_size,
                              hipStream_t stream) {
    (void)in_sizes; (void)n_in; (void)out_size; (void)ws_size;

    const float*     x  = (const float*)d_in[0];
    const long long* ei = (const long long*)d_in[1];
    const float*     W0 = (const float*)d_in[2];
    const float*     b0 = (const float*)d_in[3];
    const float*     W1 = (const float*)d_in[4];
    const float*     b1 = (const float*)d_in[5];
    const float*     W2 = (const float*)d_in[6];
    const float*     b2 = (const float*)d_in[7];
    float* out = (float*)d_out;

    float* dinv = (float*)d_ws;              // 50000 floats (padded to 50048)
    float* bufA = dinv + 50048;              // 50000*128 floats
    float* bufB = bufA + NN * 128;           // 50000*128 floats
    const long long* srcp = ei;              // edge_index[0]
    const long long* dstp = ei + NE;         // edge_index[1]

    const int N = NN, E = NE;
    dim3 blk(256);

    // deg^{-1/2}
    k_fill_ones<<<(N + 255) / 256, blk, 0, stream>>>(dinv, N);
    k_deg_accum<<<(E + 255) / 256, blk, 0, stream>>>(dstp, dinv, E);
    k_deg_rsqrt<<<(N + 255) / 256, blk, 0, stream>>>(dinv, N);

    const int gblk = (N + 127) / 128;        // 391

    // layer 0: h0 = x @ W0 ; agg0 = scatter(norm*h0[src]) + dinv^2*h0 + b0
    k_gcn_gemm<512, 128, false><<<gblk, blk, 0, stream>>>(x, W0, bufA, N);
    k_agg_init<128><<<(N * 32 + 255) / 256, blk, 0, stream>>>(bufA, dinv, b0, bufB, N);
    k_agg_edges<128, 32><<<(E * 32) / 256, blk, 0, stream>>>(srcp, dstp, dinv, bufA, bufB, E);

    // layer 1: relu folded into GEMM X load
    k_gcn_gemm<128, 128, true><<<gblk, blk, 0, stream>>>(bufB, W1, bufA, N);
    k_agg_init<128><<<(N * 32 + 255) / 256, blk, 0, stream>>>(bufA, dinv, b1, bufB, N);
    k_agg_edges<128, 32><<<(E * 32) / 256, blk, 0, stream>>>(srcp, dstp, dinv, bufA, bufB, E);

    // layer 2: final, no relu on output
    k_gcn_gemm<128, 16, true><<<gblk, blk, 0, stream>>>(bufB, W2, bufA, N);
    k_agg_init<16><<<(N * 4 + 255) / 256, blk, 0, stream>>>(bufA, dinv, b2, out, N);
    k_agg_edges<16, 4><<<(E * 4) / 256, blk, 0, stream>>>(srcp, dstp, dinv, bufA, out, E);
}